// FSCLoss3DF_48095043781221
// MI455X (gfx1250) — compile-verified
//
#include <hip/hip_runtime.h>
#include <math.h>

typedef float v2f __attribute__((ext_vector_type(2)));
typedef float v8f __attribute__((ext_vector_type(8)));

#define PI_F 3.14159265358979323846f

// ---------------------------------------------------------------------------
// One 256-point complex FFT per wave via radix-16^2 Cooley-Tukey:
//   A[n1][n2] = z[16*n1 + n2]        (16x16 complex matrix in LDS)
//   Y  = F16 * A                     (stage 1: WMMA f32 16x16x4, K chained)
//   Y[k1][n2] *= W256^(k1*n2)        (twiddle)
//   Z  = Y * F16                     (stage 2)
//   out[k1 + 16*k2] = Z[k1][k2]
// Each complex 16x16x16 matmul = 4 real matmuls = 16 v_wmma_f32_16x16x4_f32.
// sre/sim point at this wave's private 256-float LDS line.
// ---------------------------------------------------------------------------
__device__ __forceinline__ void fft256_wave(float* sre, float* sim, int lane)
{
    const int   m      = lane & 15;          // A-op row / B-op col for this lane
    const int   hi     = (lane >> 4) << 1;   // lanes 16-31 carry K+2,K+3
    const int   k1base = (lane >> 4) << 3;   // C/D rows: j or j+8
    const float w16    = -2.0f * PI_F / 16.0f;

    // ---- Stage 1: Y = F16 x A ----
    v8f yr = {0,0,0,0,0,0,0,0};
    v8f yi = {0,0,0,0,0,0,0,0};
#pragma unroll
    for (int kc = 0; kc < 4; ++kc) {
        const int k0 = 4*kc + hi;
        float s0,c0,s1,c1;
        __sincosf(w16 * (float)((m *  k0   ) & 15), &s0, &c0);
        __sincosf(w16 * (float)((m * (k0+1)) & 15), &s1, &c1);
        v2f fre  = { c0,  c1};
        v2f fim  = { s0,  s1};
        v2f fnim = {-s0, -s1};
        // B-operand chunk: A[k0..k0+1][n], n = lane&15 (16*K + n in LDS)
        v2f zr = { sre[16*k0 + m], sre[16*(k0+1) + m] };
        v2f zi = { sim[16*k0 + m], sim[16*(k0+1) + m] };
        yr = __builtin_amdgcn_wmma_f32_16x16x4_f32(false, fre,  false, zr, (short)0, yr, false, false);
        yr = __builtin_amdgcn_wmma_f32_16x16x4_f32(false, fnim, false, zi, (short)0, yr, false, false);
        yi = __builtin_amdgcn_wmma_f32_16x16x4_f32(false, fre,  false, zi, (short)0, yi, false, false);
        yi = __builtin_amdgcn_wmma_f32_16x16x4_f32(false, fim,  false, zr, (short)0, yi, false, false);
    }
    __syncthreads();

    // ---- twiddle W256^(k1*n2), store Y[k1][n2] row-major in this wave's LDS line ----
    {
        const float w256 = -2.0f * PI_F / 256.0f;
        const int n2 = m;
#pragma unroll
        for (int j = 0; j < 8; ++j) {
            const int k1 = k1base + j;
            float s, c;
            __sincosf(w256 * (float)(k1 * n2), &s, &c);
            const float re = yr[j]*c - yi[j]*s;
            const float im = yr[j]*s + yi[j]*c;
            sre[k1*16 + n2] = re;
            sim[k1*16 + n2] = im;
        }
    }
    __syncthreads();

    // ---- Stage 2: Z = Y x F16 ----
    v8f zr2 = {0,0,0,0,0,0,0,0};
    v8f zi2 = {0,0,0,0,0,0,0,0};
#pragma unroll
    for (int kc = 0; kc < 4; ++kc) {
        const int k0 = 4*kc + hi;
        // A-operand chunk: Y[m][k0..k0+1]
        v2f yre = { sre[m*16 + k0], sre[m*16 + k0 + 1] };
        v2f yim = { sim[m*16 + k0], sim[m*16 + k0 + 1] };
        float s0,c0,s1,c1;
        __sincosf(w16 * (float)(( k0    * m) & 15), &s0, &c0);
        __sincosf(w16 * (float)(((k0+1) * m) & 15), &s1, &c1);
        v2f fre  = { c0,  c1};
        v2f fim  = { s0,  s1};
        v2f fnim = {-s0, -s1};
        zr2 = __builtin_amdgcn_wmma_f32_16x16x4_f32(false, yre, false, fre,  (short)0, zr2, false, false);
        zr2 = __builtin_amdgcn_wmma_f32_16x16x4_f32(false, yim, false, fnim, (short)0, zr2, false, false);
        zi2 = __builtin_amdgcn_wmma_f32_16x16x4_f32(false, yre, false, fim,  (short)0, zi2, false, false);
        zi2 = __builtin_amdgcn_wmma_f32_16x16x4_f32(false, yim, false, fre,  (short)0, zi2, false, false);
    }
    __syncthreads();

    // ---- store Z[k1][k2] at linear index k1 + 16*k2 (k2 = m) ----
#pragma unroll
    for (int j = 0; j < 8; ++j) {
        const int k1 = k1base + j;
        sre[m*16 + k1] = zr2[j];
        sim[m*16 + k1] = zi2[j];
    }
    __syncthreads();
}

// ---------------------------------------------------------------------------
// Pass 1: pack z = ref + i*pred, FFT along W (contiguous axis).
// 131072 lines, 16 waves/block -> 8192 blocks of 512 threads.
// ---------------------------------------------------------------------------
__global__ void __launch_bounds__(512)
fsc_fft_w(const float* __restrict__ ref, const float* __restrict__ pred,
          float* __restrict__ zre, float* __restrict__ zim)
{
    __shared__ float sre[16][256];
    __shared__ float sim[16][256];
    const int  wave = threadIdx.x >> 5;
    const int  lane = threadIdx.x & 31;
    const long line = (long)blockIdx.x * 16 + wave;
    const long base = line * 256;
    float* r = sre[wave];
    float* q = sim[wave];
#pragma unroll
    for (int t = 0; t < 8; ++t) {
        const int i = lane + 32*t;
        r[i] = ref[base + i];
        q[i] = pred[base + i];
    }
    __syncthreads();
    fft256_wave(r, q, lane);
#pragma unroll
    for (int t = 0; t < 8; ++t) {
        const int i = lane + 32*t;
        zre[base + i] = r[i];
        zim[base + i] = q[i];
    }
}

// ---------------------------------------------------------------------------
// Passes 2/3: in-place FFT along a strided axis.
// line = (b:1bit)(mid:8bit)(lo:8bit); base = b*N^3 + mid*midStride + lo.
//   H axis: midStride = 65536 (d*N^2), estride = 256
//   D axis: midStride = 256   (h*N),   estride = 65536
// L2 (192 MB) holds a whole 134 MB volume -> strided traffic mostly L2-local.
// ---------------------------------------------------------------------------
__global__ void __launch_bounds__(512)
fsc_fft_strided(float* __restrict__ zre, float* __restrict__ zim,
                long midStride, long estride)
{
    __shared__ float sre[16][256];
    __shared__ float sim[16][256];
    const int  wave = threadIdx.x >> 5;
    const int  lane = threadIdx.x & 31;
    const long line = (long)blockIdx.x * 16 + wave;
    const long base = (line >> 16) * 16777216L
                    + ((line >> 8) & 255) * midStride
                    + (line & 255);
    float* r = sre[wave];
    float* q = sim[wave];
#pragma unroll
    for (int t = 0; t < 8; ++t) {
        const int i = lane + 32*t;
        r[i] = zre[base + (long)i * estride];
        q[i] = zim[base + (long)i * estride];
    }
    __syncthreads();
    fft256_wave(r, q, lane);
#pragma unroll
    for (int t = 0; t < 8; ++t) {
        const int i = lane + 32*t;
        zre[base + (long)i * estride] = r[i];
        zim[base + (long)i * estride] = q[i];
    }
}

// ---------------------------------------------------------------------------
// Pass 4: Hermitian unpack + shell accumulation.
//   F1 = (Z(k)+conj(Z(-k)))/2 ; F2 = (Z(k)-conj(Z(-k)))/(2i)
// LDS bins per block, one global f32 atomic per bin per block.
// acc layout: [b][{num,d1,d2}][129] = 774 floats.
// ---------------------------------------------------------------------------
__global__ void __launch_bounds__(256)
fsc_accum(const float* __restrict__ zre, const float* __restrict__ zim,
          float* __restrict__ acc)
{
    __shared__ float bins[774];
    for (int i = threadIdx.x; i < 774; i += blockDim.x) bins[i] = 0.0f;
    __syncthreads();

    const long TOT = 2L * 16777216L;
    for (long e = (long)blockIdx.x * blockDim.x + threadIdx.x; e < TOT;
         e += (long)gridDim.x * blockDim.x) {
        const long v  = e & 16777215L;
        const int  b  = (int)(e >> 24);
        const int  kz = (int)( v        & 255);
        const int  ky = (int)((v >> 8)  & 255);
        const int  kx = (int)( v >> 16);

        const float zr = zre[e];
        const float zi = zim[e];
        const int nx = (256 - kx) & 255;
        const int ny = (256 - ky) & 255;
        const int nz = (256 - kz) & 255;
        const long ne = ((long)b << 24) | ((long)nx << 16) | (ny << 8) | nz;
        const float cr =  zre[ne];
        const float ci = -zim[ne];

        const float Ar = 0.5f * (zr + cr);
        const float Ai = 0.5f * (zi + ci);
        const float Br =  0.5f * (zi - ci);
        const float Bi = -0.5f * (zr - cr);

        const float cross = Ar*Br + Ai*Bi;
        const float p1    = Ar*Ar + Ai*Ai;
        const float p2    = Br*Br + Bi*Bi;

        const int sx = min(kx, 256 - kx);
        const int sy = min(ky, 256 - ky);
        const int sz = min(kz, 256 - kz);
        int bin = (int)sqrtf((float)(sx*sx + sy*sy + sz*sz));
        if (bin > 128) bin = 128;

        const int o = b * 387;
        atomicAdd(&bins[o +       bin], cross);
        atomicAdd(&bins[o + 129 + bin], p1);
        atomicAdd(&bins[o + 258 + bin], p2);
    }
    __syncthreads();
    for (int i = threadIdx.x; i < 774; i += blockDim.x) atomicAdd(&acc[i], bins[i]);
}

// ---------------------------------------------------------------------------
// Final: fsc = num / sqrt(d1*d2 + eps); loss = 1 - mean(fsc^2) over 2x128.
// ---------------------------------------------------------------------------
__global__ void __launch_bounds__(128)
fsc_final(const float* __restrict__ acc, float* __restrict__ out)
{
    __shared__ float part[128];
    const int t = threadIdx.x;
    float s = 0.0f;
#pragma unroll
    for (int b = 0; b < 2; ++b) {
        const float* a = acc + b * 387;
        const float num = a[t];
        const float d1  = a[129 + t];
        const float d2  = a[258 + t];
        const float fsc = num * rsqrtf(d1 * d2 + 1e-8f);
        s += fsc * fsc;
    }
    part[t] = s;
    __syncthreads();
    for (int off = 64; off > 0; off >>= 1) {
        if (t < off) part[t] += part[t + off];
        __syncthreads();
    }
    if (t == 0) out[0] = 1.0f - part[0] / 256.0f;
}

// ---------------------------------------------------------------------------
extern "C" void kernel_launch(void* const* d_in, const int* in_sizes, int n_in,
                              void* d_out, int out_size, void* d_ws, size_t ws_size,
                              hipStream_t stream)
{
    const float* ref  = (const float*)d_in[0];
    const float* pred = (const float*)d_in[1];

    float* zre = (float*)d_ws;                  // 33,554,432 floats
    float* zim = zre + 33554432L;               // 33,554,432 floats
    float* acc = zim + 33554432L;               // 774 floats

    hipMemsetAsync(acc, 0, 774 * sizeof(float), stream);

    // 131072 lines per pass, 16 waves (lines) per 512-thread block.
    fsc_fft_w      <<<8192, 512, 0, stream>>>(ref, pred, zre, zim);
    fsc_fft_strided<<<8192, 512, 0, stream>>>(zre, zim, 65536L, 256L);   // H axis
    fsc_fft_strided<<<8192, 512, 0, stream>>>(zre, zim, 256L,   65536L); // D axis
    fsc_accum      <<<2048, 256, 0, stream>>>(zre, zim, acc);
    fsc_final      <<<1,    128, 0, stream>>>(acc, (float*)d_out);
}